// MultiScaleRoIAlign3D_88124138979452
// MI455X (gfx1250) — compile-verified
//
#include <hip/hip_runtime.h>
#include <stdint.h>

#define OUTD 7
#define TILE_ELEMS 12288   // 48 KB LDS staging tile (fits virtually all ROI patches)

typedef __attribute__((ext_vector_type(4))) unsigned int u32x4;
typedef __attribute__((ext_vector_type(8))) unsigned int u32x8;

__device__ __forceinline__ unsigned rfl(unsigned v) {
  return (unsigned)__builtin_amdgcn_readfirstlane((int)v);
}

// 4x4x4 weighted gather with precomputed element offsets per axis.
// Inlined separately for LDS-tile and global sources so address spaces infer.
__device__ __forceinline__ float gather444o(const float* __restrict__ base,
                                            const int* OZ, const float* WZ,
                                            const int* OY, const float* WY,
                                            const int* OX, const float* WX) {
  float acc = 0.0f;
#pragma unroll
  for (int p = 0; p < 4; ++p) {
    const float wp = WZ[p];
#pragma unroll
    for (int q = 0; q < 4; ++q) {
      const float wpq = wp * WY[q];
      const float* row = base + OZ[p] + OY[q];
#pragma unroll
      for (int r = 0; r < 4; ++r)
        acc = fmaf(wpq * WX[r], row[OX[r]], acc);
    }
  }
  return acc;
}

__global__ void __launch_bounds__(256)
roi_align3d_tdm_kernel(const float* __restrict__ f0, const float* __restrict__ f1,
                       const float* __restrict__ f2, const float* __restrict__ f3,
                       const float* __restrict__ boxes, float* __restrict__ out,
                       int N, int C, int cpb)
{
  __shared__ float tile[TILE_ELEMS];
  __shared__ int   sIdx[3][28];   // [axis][2*s + {lo,hi}], s = 0..13
  __shared__ float sWt[3][28];

  const int n   = blockIdx.x;
  const int tid = threadIdx.x;

  // ----- level selection (uniform per block) -----
  const float b0 = boxes[6*n+0], b1 = boxes[6*n+1], b2 = boxes[6*n+2];
  const float b3 = boxes[6*n+3], b4 = boxes[6*n+4], b5 = boxes[6*n+5];
  const float vol = (b3 - b0) * (b4 - b1) * (b5 - b2);
  const float s3  = cbrtf(vol);
  int lvl = (int)floorf(4.0f + log2f(s3 * (1.0f / 160.0f)) + 1e-6f);
  lvl = min(max(lvl, 2), 5);
  const int   L     = lvl - 2;
  const float scale = 1.0f / (float)(4 << L);
  const int   D     = 64 >> L;
  const float* feat = (L == 0) ? f0 : (L == 1) ? f1 : (L == 2) ? f2 : f3;

  // ----- per-axis sample tables (cooperative) -----
  if (tid < 42) {
    const int d = tid / 14, s = tid % 14;
    const float st   = boxes[6*n + d]     * scale;
    const float en   = boxes[6*n + 3 + d] * scale;
    const float size = fmaxf(en - st, 1.0f);
    const float bsz  = size * (1.0f / (float)OUTD);
    const float cx   = st + (0.5f * (float)s + 0.25f) * bsz;
    const float valid = (cx > -1.0f && cx < (float)D) ? 1.0f : 0.0f;
    const float ccx  = fminf(fmaxf(cx, 0.0f), (float)(D - 1));
    const int   lo   = (int)floorf(ccx);
    const int   hi   = min(lo + 1, D - 1);
    const float fr   = ccx - (float)lo;
    sIdx[d][2*s]   = lo;  sWt[d][2*s]   = (1.0f - fr) * valid;
    sIdx[d][2*s+1] = hi;  sWt[d][2*s+1] = fr * valid;
  }
  __syncthreads();

  // patch bounds (coords are monotonic in s)
  const int org0 = sIdx[0][0], org1 = sIdx[1][0], org2 = sIdx[2][0];
  const int e0 = sIdx[0][27] - org0 + 1;
  const int e1 = sIdx[1][27] - org1 + 1;
  const int e2 = sIdx[2][27] - org2 + 1;
  const long patch   = (long)e0 * e1 * e2;
  const bool useTile = (patch <= (long)TILE_ELEMS);

  const int    c0  = blockIdx.y * cpb;
  const size_t DDD = (size_t)D * D * D;
  const size_t outBase = (size_t)n * C * 343;

  // ----- per-thread voxel set (<= 2 voxels), channel-invariant tables -----
  const int vv0 = tid, vv1 = tid + 256;
  int   OZ[2][4], OY[2][4], OX[2][4];
  float WZ[2][4], WY[2][4], WX[2][4];
  {
    const int rZ = useTile ? org0 : 0, rY = useTile ? org1 : 0, rX = useTile ? org2 : 0;
    const int sZ = useTile ? e1 * e2 : D * D;
    const int sY = useTile ? e2      : D;
#pragma unroll
    for (int u = 0; u < 2; ++u) {
      const int v = (u == 0) ? vv0 : vv1;
      if (v < 343) {
        const int o2 = v % 7, o1 = (v / 7) % 7, o0 = v / 49;
#pragma unroll
        for (int k = 0; k < 4; ++k) {   // entries 4*o..4*o+3 == samples 2o,2o+1 x {lo,hi}
          OZ[u][k] = (sIdx[0][4*o0 + k] - rZ) * sZ; WZ[u][k] = sWt[0][4*o0 + k];
          OY[u][k] = (sIdx[1][4*o1 + k] - rY) * sY; WY[u][k] = sWt[1][4*o1 + k];
          OX[u][k] = (sIdx[2][4*o2 + k] - rX);      WX[u][k] = sWt[2][4*o2 + k];
        }
      }
    }
  }

  if (useTile) {
    for (int ci = 0; ci < cpb; ++ci) {
      const int c = c0 + ci;
      if (c >= C) break;

      __syncthreads();               // previous channel's LDS reads complete
      if (tid < 32) {                // wave 0 issues the TDM DMA
        const unsigned long long gaddr = (unsigned long long)(uintptr_t)
            (feat + (size_t)c * DDD + ((size_t)org0 * D + (size_t)org1) * (size_t)D + (size_t)org2);
        const unsigned ud  = (unsigned)D;
        const unsigned st1 = (unsigned)(D * D);      // tensor_dim1_stride
        const unsigned st2 = (unsigned)(D * D * D);  // tensor_dim2_stride (channel)

        u32x4 g0;
        g0[0] = rfl(1u);                                        // count=1, valid user D#
        g0[1] = rfl((unsigned)(uintptr_t)(void*)tile);          // lds_addr
        g0[2] = rfl((unsigned)(gaddr & 0xFFFFFFFFull));         // global_addr[31:0]
        g0[3] = rfl((unsigned)((gaddr >> 32) & 0x01FFFFFFull)
                    | (2u << 30));                              // addr[56:32] | type=2

        u32x8 g1;
        g1[0] = rfl(2u << 16);                                  // data_size = 4B
        g1[1] = rfl((ud & 0xFFFFu) << 16);                      // tensor_dim0 lo
        g1[2] = rfl((ud >> 16) | ((ud & 0xFFFFu) << 16));       // dim0 hi | dim1 lo
        g1[3] = rfl((ud >> 16) | ((unsigned)e2 << 16));         // dim1 hi | tile_dim0
        g1[4] = rfl((unsigned)e1 | ((unsigned)e0 << 16));       // tile_dim1 | tile_dim2
        g1[5] = rfl(ud);                                        // dim0_stride lo32
        g1[6] = rfl((st1 & 0xFFFFu) << 16);                     // s0 hi(=0) | s1 lo16
        g1[7] = rfl(st1 >> 16);                                 // s1[47:16]

        u32x4 g2;
        g2[0] = rfl(ud);                                        // tensor_dim2
        g2[1] = rfl((unsigned)C);                               // tensor_dim3
        g2[2] = rfl(st2);                                       // dim2_stride lo32
        g2[3] = rfl(0u);                                        // stride hi | tile_dim3=0

        u32x4 g3;
        g3[0] = rfl(0u); g3[1] = rfl(0u); g3[2] = rfl(0u); g3[3] = rfl(0u);

        asm volatile("tensor_load_to_lds %0, %1, %2, %3"
                     :: "s"(g0), "s"(g1), "s"(g2), "s"(g3)
                     : "memory");
        __builtin_amdgcn_s_wait_tensorcnt(0);
      }
      __syncthreads();               // patch visible in LDS to all waves

      float* __restrict__ oc = out + outBase + (size_t)c * 343;
#pragma unroll
      for (int u = 0; u < 2; ++u) {
        const int v = (u == 0) ? vv0 : vv1;
        if (v < 343) {
          const float val = gather444o(tile, OZ[u], WZ[u], OY[u], WY[u], OX[u], WX[u]);
          oc[v] = 0.125f * val;
        }
      }
    }
  } else {
    // oversized patch (rare, very skewed box): gather straight from L2-resident feat
    for (int ci = 0; ci < cpb; ++ci) {
      const int c = c0 + ci;
      if (c >= C) break;
      const float* __restrict__ src = feat + (size_t)c * DDD;
      float* __restrict__ oc = out + outBase + (size_t)c * 343;
#pragma unroll
      for (int u = 0; u < 2; ++u) {
        const int v = (u == 0) ? vv0 : vv1;
        if (v < 343) {
          const float val = gather444o(src, OZ[u], WZ[u], OY[u], WY[u], OX[u], WX[u]);
          oc[v] = 0.125f * val;
        }
      }
    }
  }
}

extern "C" void kernel_launch(void* const* d_in, const int* in_sizes, int n_in,
                              void* d_out, int out_size, void* d_ws, size_t ws_size,
                              hipStream_t stream) {
  const float* f0    = (const float*)d_in[0];
  const float* f1    = (const float*)d_in[1];
  const float* f2    = (const float*)d_in[2];
  const float* f3    = (const float*)d_in[3];
  const float* boxes = (const float*)d_in[4];
  float* out = (float*)d_out;

  const int N   = in_sizes[4] / 6;
  const int C   = in_sizes[0] / (64 * 64 * 64);
  const int cpb = (C + 3) / 4;     // channels per block (4 channel-quarters per box)

  dim3 grid((unsigned)N, 4u, 1u);  // (box, channel-quarter)
  roi_align3d_tdm_kernel<<<grid, 256, 0, stream>>>(f0, f1, f2, f3, boxes, out, N, C, cpb);
}